// AdvancedQuantumCircuitLayer_61658550502099
// MI455X (gfx1250) — compile-verified
//
#include <hip/hip_runtime.h>
#include <hip/hip_bf16.h>
#include <math.h>

#define NQ      12
#define DIM     4096      // 2^12 amplitudes
#define NLAYERS 4
#define BATCH   4096
#define FEAT    256
#define PI_F    3.14159265358979323846f

typedef __attribute__((ext_vector_type(2))) float v2f;
typedef __attribute__((ext_vector_type(8))) float v8f;

// ===========================================================================
// Compile-time GF(2) linear algebra: fold the CNOT chains into gate masks.
// Index bit p = 11-q for qubit q.  Chain permutation P: j_p = XOR_{k>=p} i_k.
// Layer-l gates act on stored pairs {a, a^mask[l][p]},
// role bit = parity(role[l][p] & a).  Expectation signs use role[4][p].
// ===========================================================================
struct GF2 { unsigned r[NQ]; };

struct Tables {
    unsigned short mask[NLAYERS + 1][NQ];   // T_l(e_p), T_l = P^{-l}
    unsigned short role[NLAYERS + 1][NQ];   // row_p(P^l)
};

constexpr GF2 gf2_identity() {
    GF2 a{};
    for (int p = 0; p < NQ; ++p) a.r[p] = 1u << p;
    return a;
}
constexpr GF2 gf2_mul(const GF2& A, const GF2& B) {   // C(x) = A(B(x))
    GF2 C{};
    for (int p = 0; p < NQ; ++p) {
        unsigned acc = 0;
        for (int k = 0; k < NQ; ++k)
            if ((A.r[p] >> k) & 1u) acc ^= B.r[k];
        C.r[p] = acc;
    }
    return C;
}
constexpr Tables make_tables() {
    GF2 P{};                                   // chain map
    for (int p = 0; p < NQ; ++p) P.r[p] = 0xFFFu & ~((1u << p) - 1u);
    GF2 Pi{};                                  // its inverse: i_p = j_p ^ j_{p+1}
    for (int p = 0; p < NQ - 1; ++p) Pi.r[p] = (1u << p) | (1u << (p + 1));
    Pi.r[NQ - 1] = 1u << (NQ - 1);

    Tables t{};
    GF2 Tl = gf2_identity();   // P^{-l}
    GF2 Rl = gf2_identity();   // P^{ l}
    for (int l = 0; l <= NLAYERS; ++l) {
        for (int p = 0; p < NQ; ++p) {
            unsigned m = 0;                    // column p of Tl
            for (int k = 0; k < NQ; ++k) m |= ((Tl.r[k] >> p) & 1u) << k;
            t.mask[l][p] = (unsigned short)m;
            t.role[l][p] = (unsigned short)Rl.r[p];
        }
        Tl = gf2_mul(Tl, Pi);
        Rl = gf2_mul(Rl, P);
    }
    return t;
}

__constant__ Tables g_tab = make_tables();

// ---------------------------------------------------------------------------
// Kernel 1: projection GEMM via V_WMMA_F32_16X16X4_F32 + tanh/normalize.
// One wave per 16-row tile; N padded 12 -> 16 via branchless zero mask.
// ---------------------------------------------------------------------------
__global__ __launch_bounds__(32) void proj_encode_kernel(
    const float* __restrict__ x,      // (4096, 256)
    const float* __restrict__ W,      // (12, 256)
    const float* __restrict__ bias,   // (12,)
    float* __restrict__ enc)          // (4096, 12) out
{
    __shared__ float yt[16][17];

    const int tile = blockIdx.x;
    const int lane = threadIdx.x;
    const int half = lane >> 4;
    const int l15  = lane & 15;
    const int row  = tile * 16 + l15;
    const int koff = half * 2;

    const float* xrow  = x + row * FEAT;
    const int    n     = l15;
    const float  wmask = (n < NQ) ? 1.f : 0.f;           // branchless N padding
    const float* wrow  = W + (n < NQ ? n : 0) * FEAT;    // clamped, always valid

    v8f c = {};
    #pragma unroll 4
    for (int k0 = 0; k0 < FEAT; k0 += 4) {
        v2f a, b;
        a.x = xrow[k0 + koff];
        a.y = xrow[k0 + koff + 1];
        b.x = wrow[k0 + koff] * wmask;
        b.y = wrow[k0 + koff + 1] * wmask;
        c = __builtin_amdgcn_wmma_f32_16x16x4_f32(
                false, a, false, b, (short)0, c, false, false);
    }

    #pragma unroll
    for (int r = 0; r < 8; ++r)
        yt[r + half * 8][l15] = c[r];
    __syncthreads();

    if (lane < 16) {
        float t[NQ];
        float ss = 0.f;
        #pragma unroll
        for (int j = 0; j < NQ; ++j) {
            float v = tanhf(yt[lane][j] + bias[j]) * PI_F;
            t[j] = v;
            ss += v * v;
        }
        float scl = PI_F * rsqrtf(ss + 1e-10f);
        const int orow = tile * 16 + lane;
        #pragma unroll
        for (int j = 0; j < NQ; ++j)
            enc[orow * NQ + j] = t[j] * scl;
    }
}

// ---------------------------------------------------------------------------
// Kernel 2: statevector circuit, one block = one sample, state in LDS.
// CNOT chains are never materialized (folded into gate index masks).
// ---------------------------------------------------------------------------
__device__ __forceinline__ float2 cmul(float2 a, float2 b) {
    return make_float2(a.x * b.x - a.y * b.y, a.x * b.y + a.y * b.x);
}
__device__ __forceinline__ float2 cadd(float2 a, float2 b) {
    return make_float2(a.x + b.x, a.y + b.y);
}

__global__ __launch_bounds__(256) void circuit_kernel(
    const float* __restrict__ enc,      // (4096, 12)
    const float* __restrict__ weights,  // (156,)
    const float* __restrict__ scaling,  // (4,)
    float* __restrict__ out)            // (4096, 12)
{
    __shared__ float2 st[DIM];                 // 32 KB statevector
    __shared__ float2 U[NLAYERS * NQ][4];      // fused RZ*RY*RX per (layer,qubit)
    __shared__ float  cosE[NQ], sinE[NQ];
    __shared__ float  fc[NQ], fs[NQ];

    const int tid = threadIdx.x;
    const int s   = blockIdx.x;

    // --- setup ---
    if (tid < NQ) {
        float e = enc[s * NQ + tid] * 0.5f;
        cosE[tid] = cosf(e);
        sinE[tid] = sinf(e);
        float wf = weights[NLAYERS * NQ * 3 + tid] * 0.5f;
        fc[tid] = cosf(wf);
        fs[tid] = sinf(wf);
    }
    if (tid < NLAYERS * NQ) {
        const int l = tid / NQ, q = tid % NQ;
        const float sc = scaling[l];
        const float ax = weights[l * 36 + q]          * sc * 0.5f;
        const float ay = weights[l * 36 + NQ + q]     * sc * 0.5f;
        const float az = weights[l * 36 + 2 * NQ + q] * sc * 0.5f;
        float cx = cosf(ax), sx = sinf(ax);
        float cy = cosf(ay), sy = sinf(ay);
        float cz = cosf(az), sz = sinf(az);
        float2 m00 = make_float2( cy * cx,  sy * sx);   // M = RY*RX
        float2 m01 = make_float2(-sy * cx, -cy * sx);
        float2 m10 = make_float2( sy * cx, -cy * sx);
        float2 m11 = make_float2( cy * cx, -sy * sx);
        float2 em  = make_float2(cz, -sz), ep = make_float2(cz, sz);
        U[tid][0] = cmul(em, m00);                      // U = RZ*M
        U[tid][1] = cmul(em, m01);
        U[tid][2] = cmul(ep, m10);
        U[tid][3] = cmul(ep, m11);
    }
    __syncthreads();

    // --- init: encoding RY layer on |0..0> is a product state ---
    {
        float rc[NQ], rs[NQ];
        #pragma unroll
        for (int q = 0; q < NQ; ++q) { rc[q] = cosE[q]; rs[q] = sinE[q]; }
        for (int i = tid; i < DIM; i += 256) {
            float amp = 1.f;
            #pragma unroll
            for (int q = 0; q < NQ; ++q)
                amp *= ((i >> (NQ - 1 - q)) & 1) ? rs[q] : rc[q];
            st[i] = make_float2(amp, 0.f);
        }
    }
    __syncthreads();

    // --- layers: fused unitaries with GF(2)-remapped index pairs ---
    for (int l = 0; l < NLAYERS; ++l) {
        for (int q = 0; q < NQ; ++q) {
            const int p = NQ - 1 - q;                 // index-bit of qubit q
            const unsigned m  = g_tab.mask[l][p];
            const unsigned rr = g_tab.role[l][p];
            const int sb = __builtin_ctz(m);          // selector bit
            const float2 u00 = U[l * NQ + q][0], u01 = U[l * NQ + q][1];
            const float2 u10 = U[l * NQ + q][2], u11 = U[l * NQ + q][3];
            #pragma unroll 2
            for (int t = tid; t < DIM / 2; t += 256) {
                const int a  = ((t >> sb) << (sb + 1)) | (t & ((1 << sb) - 1));
                const int b  = a ^ m;
                const int z  = __popc(rr & (unsigned)a) & 1;   // role of 'a'
                const int i0 = z ? b : a;             // "0"-role index
                const int i1 = z ? a : b;             // "1"-role index
                float2 s0 = st[i0], s1 = st[i1];
                st[i0] = cadd(cmul(u00, s0), cmul(u01, s1));
                st[i1] = cadd(cmul(u10, s0), cmul(u11, s1));
            }
            __syncthreads();
        }
        // CNOT chain: free (absorbed into masks of later layers)
    }

    // --- final RY layer (real 2x2), after 4 chains => use tables[4] ---
    for (int q = 0; q < NQ; ++q) {
        const int p = NQ - 1 - q;
        const unsigned m  = g_tab.mask[NLAYERS][p];
        const unsigned rr = g_tab.role[NLAYERS][p];
        const int sb = __builtin_ctz(m);
        const float cg = fc[q], sg = fs[q];
        #pragma unroll 2
        for (int t = tid; t < DIM / 2; t += 256) {
            const int a  = ((t >> sb) << (sb + 1)) | (t & ((1 << sb) - 1));
            const int b  = a ^ m;
            const int z  = __popc(rr & (unsigned)a) & 1;
            const int i0 = z ? b : a;
            const int i1 = z ? a : b;
            float2 s0 = st[i0], s1 = st[i1];
            st[i0] = make_float2(cg * s0.x - sg * s1.x, cg * s0.y - sg * s1.y);
            st[i1] = make_float2(sg * s0.x + cg * s1.x, sg * s0.y + cg * s1.y);
        }
        __syncthreads();
    }

    // --- expectation <Z_q>: sign = parity(row_p(P^4) & i) ---
    float acc[NQ];
    #pragma unroll
    for (int q = 0; q < NQ; ++q) acc[q] = 0.f;
    unsigned rrow[NQ];
    #pragma unroll
    for (int q = 0; q < NQ; ++q) rrow[q] = g_tab.role[NLAYERS][NQ - 1 - q];
    for (int i = tid; i < DIM; i += 256) {
        float2 v = st[i];
        float pr = v.x * v.x + v.y * v.y;
        #pragma unroll
        for (int q = 0; q < NQ; ++q)
            acc[q] += (__popc(rrow[q] & (unsigned)i) & 1) ? -pr : pr;
    }
    __syncthreads();                       // done reading st; reuse as scratch
    float* red = (float*)st;               // 256*12 floats
    #pragma unroll
    for (int q = 0; q < NQ; ++q)
        red[tid * NQ + q] = acc[q];
    __syncthreads();
    if (tid < NQ) {
        float sum = 0.f;
        for (int k = 0; k < 256; ++k)
            sum += red[k * NQ + tid];
        out[s * NQ + tid] = sum;
    }
}

// ---------------------------------------------------------------------------
extern "C" void kernel_launch(void* const* d_in, const int* in_sizes, int n_in,
                              void* d_out, int out_size, void* d_ws, size_t ws_size,
                              hipStream_t stream) {
    const float* x       = (const float*)d_in[0];   // (4096, 256)
    const float* W_proj  = (const float*)d_in[1];   // (12, 256)
    const float* b_proj  = (const float*)d_in[2];   // (12,)
    const float* weights = (const float*)d_in[3];   // (156,)
    const float* scaling = (const float*)d_in[4];   // (4,)
    float*       out     = (float*)d_out;           // (4096, 12)
    float*       enc     = (float*)d_ws;            // (4096, 12) scratch

    proj_encode_kernel<<<BATCH / 16, 32, 0, stream>>>(x, W_proj, b_proj, enc);
    circuit_kernel<<<BATCH, 256, 0, stream>>>(enc, weights, scaling, out);
}